// SS2D_77197742178472
// MI455X (gfx1250) — compile-verified
//
#include <hip/hip_runtime.h>
#include <hip/hip_bf16.h>
#include <cmath>

// ---- problem constants (match reference) ----
#define D_MODEL 192
#define D_INNER 384
#define D_STATE 16
#define DT_RANK 12
#define KDIR    4
#define BB      2
#define HH      64
#define WW      64
#define LL      (HH * WW)     // 4096
#define MROWS   (BB * LL)     // 8192
#define NPROJ   48            // 44 padded to 3x16
#define CPROJ   44

typedef __attribute__((ext_vector_type(2))) float v2f;
typedef __attribute__((ext_vector_type(4))) float v4f;
typedef __attribute__((ext_vector_type(8))) float v8f;

#define WMMA_F32(A, B, C) \
  __builtin_amdgcn_wmma_f32_16x16x4_f32(false, (A), false, (B), (short)0, (C), false, false)

// ------------------------------------------------------------------
// Single 16x16 tile:  OUT = sum_k A[m][k] * W[n][k]  via exact-f32 WMMA.
// ISA layout (v_wmma_f32_16x16x4_f32): A lane=M, VGPR v = K(v+2*half);
// B lane=N, VGPR v = K(v+2*half); C/D elem v = row (v+8*half), col=lane&15.
// k-blocked by 16 so all 8 loads of a block are issued before its 4 WMMAs
// (lets the scheduler use graduated s_wait_loadcnt instead of full drains).
// GUARD=true zeroes padded weight rows (only x_proj needs it).
// ------------------------------------------------------------------
template <bool GUARD>
__device__ __forceinline__ v8f wmma_tile(const float* __restrict__ arow,
                                         const float* __restrict__ brow,
                                         float bsel, int K) {
  v8f acc = {0.f, 0.f, 0.f, 0.f, 0.f, 0.f, 0.f, 0.f};
  const int half = (threadIdx.x >> 4) & 1;
  for (int k0 = 0; k0 < K; k0 += 16) {
    const int ko = k0 + 2 * half;
    v2f a[4], b[4];
#pragma unroll
    for (int j = 0; j < 4; ++j) {
      a[j] = *(const v2f*)(arow + ko + 4 * j);
      b[j] = *(const v2f*)(brow + ko + 4 * j);
    }
    if (GUARD) {
#pragma unroll
      for (int j = 0; j < 4; ++j) { b[j][0] *= bsel; b[j][1] *= bsel; }
    }
#pragma unroll
    for (int j = 0; j < 4; ++j) acc = WMMA_F32(a[j], b[j], acc);
  }
  return acc;
}

// ------------------------------------------------------------------
// GEMM with 1x4 N-tile blocking: each wave computes OUT[m0..+15][n0..+63],
// reusing one A fragment across 4 WMMAs. Inner loop is k-blocked by 16:
// 20 loads (4 A + 16 W) issued up front, then 16 WMMAs interleaved across
// the 4 accumulators -> deep load pipeline, no per-WMMA loadcnt drains.
// Requires ntiles%4==0 and no N padding. grid = Mtiles*(ntiles/4)/8.
// ------------------------------------------------------------------
__global__ void k_gemm_xw4(const float* __restrict__ X, const float* __restrict__ Wt,
                           float* __restrict__ O,
                           int lda, int ldb, int ldo, int Kdim, int n4tiles) {
  const int wave = threadIdx.x >> 5;
  const int lane = threadIdx.x & 31;
  const int half = lane >> 4, lm = lane & 15;
  const int tile = blockIdx.x * 8 + wave;
  const int nt4 = tile % n4tiles;
  const int mt  = tile / n4tiles;
  const int m0 = mt * 16, n0 = nt4 * 64;

  const float* arow = X + (size_t)(m0 + lm) * lda;
  const float* b0 = Wt + (size_t)(n0      + lm) * ldb;
  const float* b1 = Wt + (size_t)(n0 + 16 + lm) * ldb;
  const float* b2 = Wt + (size_t)(n0 + 32 + lm) * ldb;
  const float* b3 = Wt + (size_t)(n0 + 48 + lm) * ldb;

  v8f acc0 = {0.f,0.f,0.f,0.f,0.f,0.f,0.f,0.f};
  v8f acc1 = acc0, acc2 = acc0, acc3 = acc0;

  for (int k0 = 0; k0 < Kdim; k0 += 16) {
    const int ko = k0 + 2 * half;
    v2f a[4], w0[4], w1[4], w2[4], w3[4];
#pragma unroll
    for (int j = 0; j < 4; ++j) {
      const int o = ko + 4 * j;
      a[j]  = *(const v2f*)(arow + o);
      w0[j] = *(const v2f*)(b0 + o);
      w1[j] = *(const v2f*)(b1 + o);
      w2[j] = *(const v2f*)(b2 + o);
      w3[j] = *(const v2f*)(b3 + o);
    }
#pragma unroll
    for (int j = 0; j < 4; ++j) {
      acc0 = WMMA_F32(a[j], w0[j], acc0);
      acc1 = WMMA_F32(a[j], w1[j], acc1);
      acc2 = WMMA_F32(a[j], w2[j], acc2);
      acc3 = WMMA_F32(a[j], w3[j], acc3);
    }
  }
#pragma unroll
  for (int v = 0; v < 8; ++v) {
    const size_t mm = (size_t)(m0 + v + 8 * half) * ldo;
    O[mm + n0      + lm] = acc0[v];
    O[mm + n0 + 16 + lm] = acc1[v];
    O[mm + n0 + 32 + lm] = acc2[v];
    O[mm + n0 + 48 + lm] = acc3[v];
  }
}

// ------------------------------------------------------------------
// x_proj GEMM: P[k][m][c] = sum_d U[m][d] * Wxp[k][c][d]  (c<44, pad 48)
// grid = 4*512*3 / 8 = 768 blocks of 256.
// ------------------------------------------------------------------
__global__ void k_gemm_xproj(const float* __restrict__ U,
                             const float* __restrict__ Wxp,
                             float* __restrict__ P) {
  const int wave = threadIdx.x >> 5;
  const int lane = threadIdx.x & 31;
  const int half = lane >> 4, lm = lane & 15;
  const int tile = blockIdx.x * 8 + wave;        // 0..6143
  const int k = tile / (512 * 3);
  const int rem = tile % (512 * 3);
  const int nt = rem % 3, mt = rem / 3;
  const int m0 = mt * 16, n0 = nt * 16;
  const int nrow = n0 + lm;
  const float bsel = (nrow < CPROJ) ? 1.f : 0.f;
  const int nclamp = (nrow < CPROJ) ? nrow : 0;
  v8f acc = wmma_tile<true>(U + (size_t)(m0 + lm) * D_INNER,
                            Wxp + ((size_t)k * CPROJ + nclamp) * D_INNER,
                            bsel, D_INNER);
#pragma unroll
  for (int v = 0; v < 8; ++v) {
    const int mm = m0 + v + 8 * half;
    P[((size_t)k * MROWS + mm) * NPROJ + n0 + lm] = acc[v];
  }
}

// ------------------------------------------------------------------
// depthwise 3x3 conv (cross-correlation, zero pad) + bias + SiLU
// grid = B*L blocks of 384 threads (thread = channel d).
// ------------------------------------------------------------------
__global__ void k_conv(const float* __restrict__ xz, const float* __restrict__ cw,
                       const float* __restrict__ cb, float* __restrict__ U) {
  const int d = threadIdx.x;
  const int l = blockIdx.x % LL;
  const int b = blockIdx.x / LL;
  const int h = l / WW, w = l % WW;
  float acc = cb[d];
#pragma unroll
  for (int kh = 0; kh < 3; ++kh) {
    const int hh = h + kh - 1;
    if ((unsigned)hh >= (unsigned)HH) continue;
#pragma unroll
    for (int kw = 0; kw < 3; ++kw) {
      const int wc = w + kw - 1;
      if ((unsigned)wc >= (unsigned)WW) continue;
      acc += cw[d * 9 + kh * 3 + kw] *
             xz[((size_t)b * LL + hh * WW + wc) * 768 + d];
    }
  }
  acc = acc / (1.f + expf(-acc));   // silu
  U[((size_t)b * LL + l) * D_INNER + d] = acc;
}

// ------------------------------------------------------------------
// scan-order index for direction k at step t (H=W=64)
// ------------------------------------------------------------------
__device__ __forceinline__ int scan_index(int k, int t) {
  if (k == 0) return t;
  if (k == 1) return ((t & 63) << 6) | (t >> 6);
  if (k == 2) return LL - 1 - t;
  const int s = LL - 1 - t;
  return ((s & 63) << 6) | (s >> 6);
}

// ------------------------------------------------------------------
// 4-direction selective scan. One wave = 2 channels (16 states each:
// lanes 0-15 -> d0, lanes 16-31 -> d1). delta recomputed in-register
// (12-tap dt-proj + softplus). Read & write both use the natural
// spatial index, so the reference's inverse transforms are implicit;
// 4 directions + Ds*u accumulate via global f32 atomics.
// grid = B*K*(384/16) = 192 blocks of 256 (8 waves = 16 channels).
// ------------------------------------------------------------------
__global__ void k_scan(const float* __restrict__ P, const float* __restrict__ U,
                       const float* __restrict__ dtw, const float* __restrict__ dtb,
                       const float* __restrict__ Alog, const float* __restrict__ Ds,
                       float* __restrict__ Y) {
  const int wave = threadIdx.x >> 5;
  const int lane = threadIdx.x & 31;
  const int half = lane >> 4, n = lane & 15;
  const int cblk = blockIdx.x % (D_INNER / 16);          // 0..23
  const int k    = (blockIdx.x / (D_INNER / 16)) % KDIR; // 0..3
  const int b    = blockIdx.x / (KDIR * D_INNER / 16);   // 0..1
  const int d    = cblk * 16 + wave * 2 + half;
  const int kd   = k * D_INNER + d;

  const float Ad   = -expf(Alog[kd * D_STATE + n]);      // A[k][d][n]
  const float dsv  = Ds[kd];
  const float bias = dtb[kd];
  float wr[DT_RANK];
#pragma unroll
  for (int r = 0; r < DT_RANK; ++r) wr[r] = dtw[kd * DT_RANK + r];

  const float* pk = P + ((size_t)k * MROWS + (size_t)b * LL) * NPROJ;
  const float* ub = U + (size_t)b * LL * D_INNER;
  float* yb       = Y + (size_t)b * LL * D_INNER;

  float hst = 0.f;
  for (int t = 0; t < LL; ++t) {
    const int lp = scan_index(k, t);

    // prefetch a few steps ahead (strided pattern for transpose dirs)
    const int tf = (t + 4 < LL) ? t + 4 : LL - 1;
    const int lpf = scan_index(k, tf);
    __builtin_prefetch(pk + (size_t)lpf * NPROJ, 0, 1);        // global_prefetch_b8
    __builtin_prefetch(ub + (size_t)lpf * D_INNER + d, 0, 1);

    const float* pr = pk + (size_t)lp * NPROJ;
    const v4f p0 = *(const v4f*)(pr);
    const v4f p1 = *(const v4f*)(pr + 4);
    const v4f p2 = *(const v4f*)(pr + 8);
    float dlt = bias;
    dlt += p0[0]*wr[0] + p0[1]*wr[1] + p0[2]*wr[2]  + p0[3]*wr[3];
    dlt += p1[0]*wr[4] + p1[1]*wr[5] + p1[2]*wr[6]  + p1[3]*wr[7];
    dlt += p2[0]*wr[8] + p2[1]*wr[9] + p2[2]*wr[10] + p2[3]*wr[11];
    dlt = (dlt > 20.f) ? dlt : log1pf(expf(dlt));        // softplus

    const float Bn = pr[DT_RANK + n];
    const float Cn = pr[DT_RANK + D_STATE + n];
    const float uu = ub[(size_t)lp * D_INNER + d];

    hst = expf(dlt * Ad) * hst + (dlt * uu) * Bn;
    float yv = hst * Cn;
    yv += __shfl_xor(yv, 1, 32);
    yv += __shfl_xor(yv, 2, 32);
    yv += __shfl_xor(yv, 4, 32);
    yv += __shfl_xor(yv, 8, 32);                         // sum over 16 states
    if (n == 0)
      atomicAdd(&yb[(size_t)lp * D_INNER + d], yv + dsv * uu);
  }
}

// ------------------------------------------------------------------
// LayerNorm over D_INNER + gate by silu(z); 1 wave per row (12 vals/lane)
// grid = 8192/8 = 1024 blocks of 256.
// ------------------------------------------------------------------
__global__ void k_ln(const float* __restrict__ Y, const float* __restrict__ xz,
                     const float* __restrict__ g, const float* __restrict__ be,
                     float* __restrict__ O) {
  const int wave = threadIdx.x >> 5;
  const int lane = threadIdx.x & 31;
  const size_t m = (size_t)blockIdx.x * 8 + wave;
  const float* yr = Y + m * D_INNER;
  float v[12], s = 0.f, s2 = 0.f;
#pragma unroll
  for (int i = 0; i < 12; ++i) {
    const float x = yr[lane + i * 32];
    v[i] = x; s += x; s2 += x * x;
  }
#pragma unroll
  for (int o = 1; o < 32; o <<= 1) {
    s  += __shfl_xor(s,  o, 32);
    s2 += __shfl_xor(s2, o, 32);
  }
  const float mean = s * (1.f / D_INNER);
  const float var  = s2 * (1.f / D_INNER) - mean * mean;
  const float rstd = rsqrtf(var + 1e-5f);
  const float* zr = xz + m * 768 + D_INNER;
  float* orow = O + m * D_INNER;
#pragma unroll
  for (int i = 0; i < 12; ++i) {
    const int dd = lane + i * 32;
    const float zz = zr[dd];
    const float yn = (v[i] - mean) * rstd * g[dd] + be[dd];
    orow[dd] = yn * (zz / (1.f + expf(-zz)));
  }
}

// ------------------------------------------------------------------
extern "C" void kernel_launch(void* const* d_in, const int* in_sizes, int n_in,
                              void* d_out, int out_size, void* d_ws, size_t ws_size,
                              hipStream_t stream) {
  (void)in_sizes; (void)n_in; (void)out_size; (void)ws_size;
  const float* x    = (const float*)d_in[0];
  const float* wi   = (const float*)d_in[1];
  const float* cw   = (const float*)d_in[2];
  const float* cb   = (const float*)d_in[3];
  const float* wxp  = (const float*)d_in[4];
  const float* dtw  = (const float*)d_in[5];
  const float* dtb  = (const float*)d_in[6];
  const float* alog = (const float*)d_in[7];
  const float* ds   = (const float*)d_in[8];
  const float* g    = (const float*)d_in[9];
  const float* be   = (const float*)d_in[10];
  const float* wo   = (const float*)d_in[11];
  float* out = (float*)d_out;

  float* xz   = (float*)d_ws;                           // (8192, 768)
  float* u    = xz   + (size_t)MROWS * 768;             // (8192, 384)
  float* proj = u    + (size_t)MROWS * D_INNER;         // (4, 8192, 48)
  float* y    = proj + (size_t)KDIR * MROWS * NPROJ;    // (8192, 384)
  float* ynz  = y    + (size_t)MROWS * D_INNER;         // (8192, 384)

  hipMemsetAsync(y, 0, (size_t)MROWS * D_INNER * sizeof(float), stream);

  // in_proj: xz(8192x768) = x(8192x192) @ Wi(768x192)^T ; 512 * 12 wave-tiles
  k_gemm_xw4<<<768, 256, 0, stream>>>(x, wi, xz, 192, 192, 768, 192, 12);
  // depthwise conv + SiLU
  k_conv<<<MROWS, D_INNER, 0, stream>>>(xz, cw, cb, u);
  // x_proj (4 GEMMs, N=44 padded to 48)
  k_gemm_xproj<<<768, 256, 0, stream>>>(u, wxp, proj);
  // 4-direction selective scan with fused dt-proj/softplus + Ds term
  k_scan<<<BB * KDIR * (D_INNER / 16), 256, 0, stream>>>(proj, u, dtw, dtb,
                                                         alog, ds, y);
  // LayerNorm + silu(z) gating
  k_ln<<<MROWS / 8, 256, 0, stream>>>(y, xz, g, be, ynz);
  // out_proj: out(8192x192) = ynz(8192x384) @ Wo(192x384)^T ; 512 * 3 wave-tiles
  k_gemm_xw4<<<192, 256, 0, stream>>>(ynz, wo, out, 384, 384, 192, 384, 3);
}